// SphericalVectorPool_15788299780325
// MI455X (gfx1250) — compile-verified
//
#include <hip/hip_runtime.h>
#include <hip/hip_bf16.h>

// CDNA5 / gfx1250: wave32, WMMA 16x16x4 f32 (full-precision matrix pipe).
typedef __attribute__((ext_vector_type(2))) float v2f;
typedef __attribute__((ext_vector_type(4))) float v4f;
typedef __attribute__((ext_vector_type(8))) float v8f;

#define WAVES_PER_BLOCK 8
#define TILE_N 32              // sources staged per wave per tile
#define DF_FLOATS 64           // 32 x (d, f)
#define U_FLOATS 144           // 32 x (1, ux, uy, uz) + 4 rows of slack for don't-care lane reads
#define LDSF_PER_WAVE (DF_FLOATS + U_FLOATS)

__global__ __launch_bounds__(WAVES_PER_BLOCK * 32)
void SphericalVectorPool_kernel(const float* __restrict__ f,
                                const float* __restrict__ coords,
                                const float* __restrict__ out_coords,
                                const float* __restrict__ mu,
                                const float* __restrict__ r_norms,
                                const float* __restrict__ a_norm_0,
                                const float* __restrict__ a_norm_1,
                                float* __restrict__ out,
                                int N, int M, int total_bm) {
    __shared__ float smem[WAVES_PER_BLOCK * LDSF_PER_WAVE];

    const int tid  = threadIdx.x;
    const int lane = tid & 31;
    const int wid  = tid >> 5;
    const int g    = blockIdx.x * WAVES_PER_BLOCK + wid;   // flat (b, m), uniform per wave
    if (g >= total_bm) return;                             // uniform branch: EXEC stays full
    const int b = g / M;

    float* dfb = smem + wid * LDSF_PER_WAVE;   // per-wave private: (d, f) pairs, n-indexed
    float* ub  = dfb + DF_FLOATS;              // per-wave private: (1, ux, uy, uz) rows

    // Anchor R (uniform per wave -> scalarized by compiler)
    const float Rx = out_coords[(size_t)g * 3 + 0];
    const float Ry = out_coords[(size_t)g * 3 + 1];
    const float Rz = out_coords[(size_t)g * 3 + 2];

    // Lane's radial basis: k = lane & 15 (WMMA B columns / D columns)
    const int   k    = lane & 15;
    const float mul2 = mu[k] * 1.4426950408889634f;   // mu * log2(e): exp(-mu d) = exp2(-mul2 d)

    v8f acc = {0.f, 0.f, 0.f, 0.f, 0.f, 0.f, 0.f, 0.f};   // D rows 0..3 used: c0, cv_x, cv_y, cv_z

    const int    numTiles = N >> 5;
    const size_t srcBase  = (size_t)b * N;
    // A/B operand K-slot selector: lanes 0-15 hold contraction slots {0,1}, lanes 16-31 hold {2,3}
    const int half = (lane >> 4) << 1;

    for (int tile = 0; tile < numTiles; ++tile) {
        const int    n  = tile * TILE_N + lane;
        const size_t ci = (srcBase + n) * 3;

        if (tile + 1 < numTiles) {
            __builtin_prefetch(coords + ci + TILE_N * 3, 0, 1);   // global_prefetch_b8
        }

        const float cx = coords[ci + 0];
        const float cy = coords[ci + 1];
        const float cz = coords[ci + 2];
        const float fv = f[srcBase + n];

        const float dx = cx - Rx, dy = cy - Ry, dz = cz - Rz;
        const float sq  = dx * dx + dy * dy + dz * dz;
        const float d   = __builtin_amdgcn_sqrtf(sq);
        const float inv = __builtin_amdgcn_rsqf(sq);

        // Stage this tile's geometry for WMMA-layout consumption (lane <-> n here).
        dfb[lane * 2 + 0] = d;
        dfb[lane * 2 + 1] = fv;
        ub[lane * 4 + 0] = 1.0f;        // row c=0 weight
        ub[lane * 4 + 1] = dx * inv;    // row c=1: u_x
        ub[lane * 4 + 2] = dy * inv;    // row c=2: u_y
        ub[lane * 4 + 3] = dz * inv;    // row c=3: u_z

        // Intra-wave LDS store->load hazard + compiler barrier.
        asm volatile("s_wait_dscnt 0" ::: "memory");

        // 8 WMMA chunks of 4 sources: out[c,k] += sum_n G[c,n] * W[n,k]
        #pragma unroll
        for (int ch = 0; ch < 8; ++ch) {
            const int n0 = ch * 4 + half;                     // this lane's 2 contraction slots
            const v4f df = *(const v4f*)(dfb + n0 * 2);       // d0, f0, d1, f1 (16B aligned bcast)

            // B[n, k] = exp(-mu_k * d_n) * f_n  (lane = column k; 2 K-slots per lane)
            v2f bm;
            bm.x = __builtin_amdgcn_exp2f(-mul2 * df.x) * df.y;
            bm.y = __builtin_amdgcn_exp2f(-mul2 * df.z) * df.w;

            // A[c, n]: lane = row c (rows >=4 read don't-care in-bounds data; D rows 4..15 unread)
            v2f am;
            am.x = ub[n0 * 4 + k];
            am.y = ub[(n0 + 1) * 4 + k];

            // v_wmma_f32_16x16x4_f32: full-f32 MAC + free cross-lane reduction over sources
            acc = __builtin_amdgcn_wmma_f32_16x16x4_f32(
                false, am, false, bm, (short)0, acc, false, false);
        }
    }

    // D layout: VGPR r, lanes 0-15 -> (row=r, col=lane). Rows 0..3 = [c0, cv_x, cv_y, cv_z].
    if (lane < 16) {
        const float rn = r_norms[lane];
        const float s0 = rn * a_norm_0[0];
        const float s1 = rn * a_norm_1[0];
        float* o = out + (size_t)g * 64;
        o[lane +  0] = acc[0] * s0;
        o[lane + 16] = acc[1] * s1;
        o[lane + 32] = acc[2] * s1;
        o[lane + 48] = acc[3] * s1;
    }
}

extern "C" void kernel_launch(void* const* d_in, const int* in_sizes, int n_in,
                              void* d_out, int out_size, void* d_ws, size_t ws_size,
                              hipStream_t stream) {
    const float* f          = (const float*)d_in[0];
    const float* coords     = (const float*)d_in[1];
    const float* out_coords = (const float*)d_in[2];
    const float* mu         = (const float*)d_in[3];
    const float* r_norms    = (const float*)d_in[4];
    const float* a_norm_0   = (const float*)d_in[5];
    const float* a_norm_1   = (const float*)d_in[6];
    float* out = (float*)d_out;

    const int B  = 2;
    const int BN = in_sizes[0];          // B * N  (f is [B, N])
    const int BM = in_sizes[2] / 3;      // B * M  (out_coords is [B, M, 3])
    const int N  = BN / B;
    const int M  = BM / B;

    const int totalWaves = BM;           // one wave per (b, m)
    const int blocks = (totalWaves + WAVES_PER_BLOCK - 1) / WAVES_PER_BLOCK;

    SphericalVectorPool_kernel<<<blocks, WAVES_PER_BLOCK * 32, 0, stream>>>(
        f, coords, out_coords, mu, r_norms, a_norm_0, a_norm_1, out, N, M, totalWaves);
}